// CrossAttentionModule_8615704396324
// MI455X (gfx1250) — compile-verified
//
#include <hip/hip_runtime.h>
#include <math.h>

// Problem shape fixed by the reference.
static constexpr int B  = 2;
static constexpr int S  = 2048;
static constexpr int A  = 16;
static constexpr int H  = 1024;
static constexpr int NH = 16;
static constexpr int HD = H / NH;   // 64

typedef __attribute__((ext_vector_type(2))) float v2f;
typedef __attribute__((ext_vector_type(8))) float v8f;

// ---------------------------------------------------------------------------
// Kernel 1: h2_agg (masked mean over aspects) -> q = h2_agg @ Wq^T ->
//           wq_eff[b,n,h] = scale * sum_d q[n*HD+d] * Wk[n*HD+d, h]
// One block per batch, 1024 threads. All tiny work; keeps everything in LDS.
// ---------------------------------------------------------------------------
__global__ void prep_kernel(const float* __restrict__ h2,
                            const unsigned char* __restrict__ amask,
                            const float* __restrict__ Wq,
                            const float* __restrict__ Wk,
                            float* __restrict__ wq_eff) {
    const int b   = blockIdx.x;
    const int tid = threadIdx.x;              // 0..1023 == H
    __shared__ float agg[H];
    __shared__ float q[H];

    // --- masked mean over aspects (each thread owns one hidden channel) ---
    {
        float cnt = 0.0f, sum = 0.0f;
        for (int a = 0; a < A; ++a) {
            const float mv = amask[b * A + a] ? 1.0f : 0.0f;
            cnt += mv;
            sum += mv * h2[((size_t)b * A + a) * H + tid];
        }
        agg[tid] = sum / fmaxf(cnt, 1.0f);
    }
    __syncthreads();

    // --- q[o] = sum_h agg[h] * Wq[o,h]  (torch y = x @ W^T) ---
    {
        const float* wrow = Wq + (size_t)tid * H;
        float acc = 0.0f;
        for (int h = 0; h < H; ++h) acc += agg[h] * wrow[h];
        q[tid] = acc;
    }
    __syncthreads();

    // --- wq_eff[b,n,h] = scale * sum_d q[n*HD+d] * Wk[n*HD+d, h] ---
    const float scale = rsqrtf((float)HD);    // hd^-0.5
    for (int idx = tid; idx < NH * H; idx += H) {
        const int n = idx / H;
        const int h = idx % H;
        float acc = 0.0f;
        for (int d = 0; d < HD; ++d)
            acc += q[n * HD + d] * Wk[(size_t)(n * HD + d) * H + h];
        wq_eff[((size_t)b * NH + n) * H + h] = acc * scale;
    }
}

// ---------------------------------------------------------------------------
// Kernel 2: scores[b,k,n] = h1[b] (SxH) @ wq_eff[b]^T (Hx16) via
//           V_WMMA_F32_16X16X4_F32. One wave per 16-row tile, 8 waves/block.
//   A (16x4 f32): lanes 0-15 hold M=lane, VGPR0=K0/VGPR1=K1;
//                 lanes 16-31 hold M=lane-16, K2/K3  -> float2 per lane.
//   B (4x16 f32): lane = column N (head), same half-wave K split.
//   C/D (16x16):  lane<16 -> N=lane, acc[i] = row M=i; lane>=16 -> M=i+8.
// ---------------------------------------------------------------------------
__global__ void scores_wmma_kernel(const float* __restrict__ h1,
                                   const float* __restrict__ wq_eff,
                                   float* __restrict__ scores) {
    const int lane = threadIdx.x & 31;
    const int wave = threadIdx.x >> 5;                 // 0..7
    const int tile = blockIdx.x * 8 + wave;            // 0..B*S/16-1
    const int b    = tile / (S / 16);
    const int m0   = (tile % (S / 16)) * 16;

    const int half = lane >> 4;                        // 0: K{0,1}  1: K{2,3}
    const int idx  = lane & 15;                        // A: row M; B: head N

    const float* Abase = h1     + ((size_t)b * S  + m0 + idx) * H + half * 2;
    const float* Bbase = wq_eff + ((size_t)b * NH + idx)      * H + half * 2;

    v8f acc = {};
    #pragma unroll 8
    for (int kk = 0; kk < H; kk += 4) {
        v2f a  = *(const v2f*)(Abase + kk);
        v2f bb = *(const v2f*)(Bbase + kk);
        // D = A x B + C : 8 args (neg_a, A, neg_b, B, c_mod, C, reuse_a, reuse_b)
        acc = __builtin_amdgcn_wmma_f32_16x16x4_f32(
            false, a, false, bb, (short)0, acc, false, false);
    }

    // Write back: lane owns head n=idx; acc[i] is sentence row m0 + i (+8 for hi half)
    const int mbase = m0 + (half ? 8 : 0);
    #pragma unroll
    for (int i = 0; i < 8; ++i)
        scores[((size_t)b * S + mbase + i) * NH + idx] = acc[i];
}

// ---------------------------------------------------------------------------
// Kernel 3: per-batch masked softmax over k (per head) + mean over heads.
//   w_mean[b,k] = (1/NH) * sum_n exp(s[b,k,n]-mx[n]) / Z[n]   (0 if masked)
// One block per batch, 256 threads (16 per head).
// ---------------------------------------------------------------------------
__global__ void softmax_mean_kernel(const float* __restrict__ scores,
                                    const unsigned char* __restrict__ smask,
                                    float* __restrict__ wmean) {
    const int b   = blockIdx.x;
    const int tid = threadIdx.x;      // 0..255
    const int n   = tid & 15;         // head
    const int g   = tid >> 4;         // 0..15: 16 workers per head

    __shared__ float red[NH][17];     // padded to dodge bank conflicts
    __shared__ float mx[NH], inv[NH];

    // --- pass 1: per-head max over valid k ---
    float m = -INFINITY;
    for (int k = g; k < S; k += 16) {
        if (smask[b * S + k]) {
            const float v = scores[((size_t)b * S + k) * NH + n];
            m = fmaxf(m, v);
        }
    }
    red[n][g] = m;
    __syncthreads();
    if (g == 0) {
        float mm = red[n][0];
        for (int i = 1; i < 16; ++i) mm = fmaxf(mm, red[n][i]);
        mx[n] = mm;
    }
    __syncthreads();

    // --- pass 2: per-head sum of exp ---
    const float mm = mx[n];
    float s = 0.0f;
    for (int k = g; k < S; k += 16) {
        if (smask[b * S + k])
            s += __expf(scores[((size_t)b * S + k) * NH + n] - mm);
    }
    red[n][g] = s;
    __syncthreads();
    if (g == 0) {
        float ss = 0.0f;
        for (int i = 0; i < 16; ++i) ss += red[n][i];
        inv[n] = 1.0f / ss;
    }
    __syncthreads();

    // --- pass 3: head-mean weight per key position ---
    for (int k = tid; k < S; k += 256) {
        if (!smask[b * S + k]) { wmean[b * S + k] = 0.0f; continue; }
        const float* srow = scores + ((size_t)b * S + k) * NH;
        float acc = 0.0f;
        #pragma unroll
        for (int nn = 0; nn < NH; ++nn)
            acc += __expf(srow[nn] - mx[nn]) * inv[nn];
        wmean[b * S + k] = acc * (1.0f / (float)NH);
    }
}

// ---------------------------------------------------------------------------
// Kernel 4: out[b,q,:] = w_mean[b,:]  — the bandwidth-bound broadcast store
// (33.5 MB). One block per output row, float4 stream writes; the 8 KB source
// row stays hot in cache.
// ---------------------------------------------------------------------------
__global__ void broadcast_rows_kernel(const float* __restrict__ wmean,
                                      float* __restrict__ out) {
    const int row = blockIdx.x;            // b*S + q
    const int b   = row / S;
    const float4* __restrict__ src = (const float4*)(wmean + (size_t)b * S);
    float4* __restrict__ dst = (float4*)(out + (size_t)row * S);
    #pragma unroll 2
    for (int i = threadIdx.x; i < S / 4; i += 256)
        dst[i] = src[i];
}

// ---------------------------------------------------------------------------
extern "C" void kernel_launch(void* const* d_in, const int* in_sizes, int n_in,
                              void* d_out, int out_size, void* d_ws, size_t ws_size,
                              hipStream_t stream) {
    (void)in_sizes; (void)n_in; (void)out_size; (void)ws_size;

    const float*         h1    = (const float*)d_in[0];          // [B,S,H]
    const float*         h2    = (const float*)d_in[1];          // [B,A,H]
    const unsigned char* smask = (const unsigned char*)d_in[2];  // [B,S] bool
    const unsigned char* amask = (const unsigned char*)d_in[3];  // [B,A] bool
    const float*         Wq    = (const float*)d_in[4];          // [H,H]
    const float*         Wk    = (const float*)d_in[5];          // [H,H]
    float*               out   = (float*)d_out;                  // [B,S,S]

    char* ws = (char*)d_ws;
    float* wq_eff = (float*)ws;                                        // B*NH*H
    float* scores = (float*)(ws + (size_t)B * NH * H * sizeof(float)); // B*S*NH
    float* wmean  = (float*)(ws + (size_t)B * NH * H * sizeof(float)
                                + (size_t)B * S * NH * sizeof(float)); // B*S

    prep_kernel<<<B, H, 0, stream>>>(h2, amask, Wq, Wk, wq_eff);
    scores_wmma_kernel<<<(B * S / 16) / 8, 256, 0, stream>>>(h1, wq_eff, scores);
    softmax_mean_kernel<<<B, 256, 0, stream>>>(scores, smask, wmean);
    broadcast_rows_kernel<<<B * S, 256, 0, stream>>>(wmean, out);
}